// LightSS2D_79336635891833
// MI455X (gfx1250) — compile-verified
//
#include <hip/hip_runtime.h>
#include <hip/hip_bf16.h>
#include <math.h>

// ---------------------------------------------------------------------------
// LightSS2D (Mamba-style selective scan over flattened H*W) for MI455X/gfx1250
//   - f32 WMMA (V_WMMA_F32_16X16X4_F32) for the three projections
//   - W tiles staged to LDS via GLOBAL_LOAD_ASYNC_TO_LDS (ASYNCcnt) builtin;
//     bank-conflict-free padded rows; 2 M-tiles per wave sharing one B frag
//   - chunked parallel scan (192 chunks x 192 channels, 16 states in regs)
// ---------------------------------------------------------------------------

typedef __attribute__((ext_vector_type(2))) float v2f;
typedef __attribute__((ext_vector_type(8))) float v8f;
typedef __attribute__((ext_vector_type(4))) int   v4i;
typedef v4i __attribute__((address_space(1))) gv4i;   // global 16B vector
typedef v4i __attribute__((address_space(3))) lv4i;   // LDS    16B vector

#define BQ      2
#define HQ      96
#define WQ      96
#define DMODEL  96
#define LSEQ    (HQ * WQ)          // 9216
#define DINNER  192
#define DXZ     (2 * DINNER)       // 384
#define DSTATE  16
#define DTRANK  6
#define DBC_DIM 38                 // dt_rank + 2*d_state
#define DBC_PAD 48                 // padded stride for dbc
#define MTOT    (BQ * LSEQ)        // 18432
#define LC      96                 // scan chunk length
#define NCH     (LSEQ / LC)        // 96 chunks per batch

#define LDS_PAD 4                  // per-row pad -> conflict-free B reads
#define LDS_MAX 9408               // 48 * (192+4) floats (worst case)

// 16-byte global -> LDS copy; async (ASYNCcnt-tracked) path if available.
__device__ __forceinline__ void copy16_g2lds(float* lds_dst, const float* g_src)
{
#if defined(__has_builtin)
#if __has_builtin(__builtin_amdgcn_global_load_async_to_lds_b128)
    __builtin_amdgcn_global_load_async_to_lds_b128((gv4i*)g_src,
                                                   (lv4i*)lds_dst, 0, 0);
    return;
#endif
#endif
    *(float4*)lds_dst = *(const float4*)g_src;
}

__device__ __forceinline__ void wait_async_lds()
{
#if defined(__has_builtin)
#if __has_builtin(__builtin_amdgcn_s_wait_asynccnt)
    __builtin_amdgcn_s_wait_asynccnt(0);
    return;
#endif
#endif
    asm volatile("s_wait_asynccnt 0x0" ::: "memory");
}

// ---------------------------------------------------------------------------
// f32 WMMA GEMM:  C[M,N] = A[M,K] * W[N,K]^T   (all row-major)
// Block = 2 (M) x wavesN (N) waves; each wave computes TWO 16x16 M-tiles that
// share one B fragment. W tile for the block's N-range staged in LDS.
// V_WMMA_F32_16X16X4_F32 lane layout (ISA 7.12.2):
//   A 16x4 : lanes 0-15 -> row M=lane, {K=k0,k0+1}; lanes 16-31 -> {k0+2,k0+3}
//   B 4x16 : symmetric half-split over K, N = lane&15
//   C/D    : VGPR r -> (M = r + 8*(lane>=16), N = lane&15)
// ---------------------------------------------------------------------------
__global__ void gemm_wmma_f32(const float* __restrict__ A, int lda,
                              const float* __restrict__ W, int ldw, int nvalid,
                              float* __restrict__ C, int ldc, int K)
{
    __shared__ float ldsW[LDS_MAX];

    const int lane   = threadIdx.x & 31;
    const int wave   = threadIdx.x >> 5;
    const int wavesN = (blockDim.x >> 5) >> 1;     // waves arranged 2 x wavesN
    const int waveM  = wave / wavesN;
    const int waveN  = wave % wavesN;
    const int mBase  = blockIdx.x * 64 + waveM * 32;   // 2 tiles: mBase, mBase+16
    const int nBase0 = blockIdx.y * (wavesN * 16);     // block N origin
    const int nBase  = nBase0 + waveN * 16;
    const int half   = lane >> 4;                  // K half select
    const int l15    = lane & 15;
    const int ldsK   = K + LDS_PAD;

    // ---- stage W[nBase0 .. nBase0+Nblock) x K into LDS (zero-padded rows) ---
    {
        const int Nblock = wavesN * 16;
        const int cpr    = K >> 2;                 // 16B chunks per row
        const int total  = Nblock * cpr;
        for (int i = threadIdx.x; i < total; i += blockDim.x) {
            const int r = i / cpr;
            const int c4 = (i - r * cpr) << 2;
            float* dst = &ldsW[r * ldsK + c4];
            if (nBase0 + r < nvalid) {
                copy16_g2lds(dst, W + (size_t)(nBase0 + r) * ldw + c4);
            } else {
                float4 z = {0.f, 0.f, 0.f, 0.f};
                *(float4*)dst = z;
            }
        }
        wait_async_lds();
        __syncthreads();
    }

    const float* Ap0 = A + (size_t)(mBase + l15) * lda;        // tile 0
    const float* Ap1 = Ap0 + (size_t)16 * lda;                 // tile 1
    const float* Bl  = &ldsW[(waveN * 16 + l15) * ldsK];       // lane's W row

    v8f acc0 = {};
    v8f acc1 = {};
    for (int k0 = 0; k0 < K; k0 += 4) {
        const int ka = k0 + half * 2;
        v2f a0, a1, b;
        b.x  = Bl[ka];
        b.y  = Bl[ka + 1];
        a0.x = Ap0[ka];
        a0.y = Ap0[ka + 1];
        a1.x = Ap1[ka];
        a1.y = Ap1[ka + 1];
        acc0 = __builtin_amdgcn_wmma_f32_16x16x4_f32(false, a0, false, b,
                                                     (short)0, acc0, false, false);
        acc1 = __builtin_amdgcn_wmma_f32_16x16x4_f32(false, a1, false, b,
                                                     (short)0, acc1, false, false);
    }

    const int nCol = nBase + l15;
    if (nCol < nvalid) {
#pragma unroll
        for (int r = 0; r < 8; ++r) {
            const int row = mBase + half * 8 + r;
            C[(size_t)row * ldc + nCol]        = acc0[r];
            C[(size_t)(row + 16) * ldc + nCol] = acc1[r];
        }
    }
}

// ---------------------------------------------------------------------------
// Causal depthwise conv (k=3) over L + SiLU.  xs = xz[:, :, 0:192]
// ---------------------------------------------------------------------------
__global__ void conv_silu_kernel(const float* __restrict__ xz,
                                 const float* __restrict__ conv_w,
                                 const float* __restrict__ conv_b,
                                 float* __restrict__ xc)
{
    const int idx = blockIdx.x * blockDim.x + threadIdx.x;
    if (idx >= MTOT * DINNER) return;
    const int d = idx % DINNER;
    const int m = idx / DINNER;
    const int l = m % LSEQ;

    float s = conv_b[d];
#pragma unroll
    for (int k = 0; k < 3; ++k) {
        const int lp = l + k - 2;
        if (lp >= 0)
            s += xz[(size_t)(m + k - 2) * DXZ + d] * conv_w[d * 3 + k];
    }
    const float sig = 1.0f / (1.0f + __expf(-s));
    xc[(size_t)m * DINNER + d] = s * sig;
}

// ---------------------------------------------------------------------------
// delta = softplus(dt @ W_dt^T + b_dt), dt = dbc[:, 0:6]
// ---------------------------------------------------------------------------
__global__ void delta_kernel(const float* __restrict__ dbc,
                             const float* __restrict__ W_dt,
                             const float* __restrict__ b_dt,
                             float* __restrict__ delta)
{
    const int idx = blockIdx.x * blockDim.x + threadIdx.x;
    if (idx >= MTOT * DINNER) return;
    const int d = idx % DINNER;
    const int m = idx / DINNER;

    float s = b_dt[d];
#pragma unroll
    for (int r = 0; r < DTRANK; ++r)
        s += dbc[(size_t)m * DBC_PAD + r] * W_dt[d * DTRANK + r];
    const float sp = (s > 20.0f) ? s : log1pf(__expf(s));
    delta[(size_t)m * DINNER + d] = sp;
}

// ---------------------------------------------------------------------------
// Scan phase 1: per-chunk local scan (h0 = 0) -> cumulative decay P, local end h
// grid.x = B*NCH chunks, blockDim = 192 (one thread per channel, 16 states/regs)
// ---------------------------------------------------------------------------
__global__ void scan_pass1(const float* __restrict__ delta,
                           const float* __restrict__ dbc,
                           const float* __restrict__ xc,
                           const float* __restrict__ A_log,
                           float* __restrict__ Pout,
                           float* __restrict__ Hend)
{
    const int chunk = blockIdx.x;
    const int d = threadIdx.x;
    const int b = chunk / NCH;
    const int c = chunk % NCH;
    const int l0 = c * LC;

    float Av[DSTATE], P[DSTATE], h[DSTATE];
#pragma unroll
    for (int n = 0; n < DSTATE; ++n) {
        Av[n] = -__expf(A_log[d * DSTATE + n]);
        P[n] = 1.0f;
        h[n] = 0.0f;
    }

    for (int l = l0; l < l0 + LC; ++l) {
        const size_t m = (size_t)b * LSEQ + l;
        __builtin_prefetch(delta + (m + 1) * DINNER + d, 0, 0);
        const float dl = delta[m * DINNER + d];
        const float xv = xc[m * DINNER + d];
        const float* Bm = dbc + m * DBC_PAD + DTRANK;
        const float dlx = dl * xv;
#pragma unroll
        for (int n = 0; n < DSTATE; ++n) {
            const float a = __expf(dl * Av[n]);
            P[n] *= a;
            h[n] = fmaf(a, h[n], dlx * Bm[n]);
        }
    }

    const int base = chunk * (DINNER * DSTATE) + d * DSTATE;
#pragma unroll
    for (int n = 0; n < DSTATE; ++n) {
        Pout[base + n] = P[n];
        Hend[base + n] = h[n];
    }
}

// ---------------------------------------------------------------------------
// Scan phase 2: sequential combine over chunk boundaries (NCH steps)
// ---------------------------------------------------------------------------
__global__ void scan_combine(const float* __restrict__ P,
                             const float* __restrict__ Hend,
                             float* __restrict__ Hinit)
{
    const int t = blockIdx.x * blockDim.x + threadIdx.x;
    if (t >= BQ * DINNER * DSTATE) return;
    const int b = t / (DINNER * DSTATE);
    const int dn = t % (DINNER * DSTATE);

    float h = 0.0f;
    for (int c = 0; c < NCH; ++c) {
        const int off = (b * NCH + c) * (DINNER * DSTATE) + dn;
        Hinit[off] = h;
        h = fmaf(P[off], h, Hend[off]);
    }
}

// ---------------------------------------------------------------------------
// Scan phase 3: re-scan with correct h_init, fuse y = h.C + xc*D, gate silu(z)
// ---------------------------------------------------------------------------
__global__ void scan_pass2(const float* __restrict__ delta,
                           const float* __restrict__ dbc,
                           const float* __restrict__ xc,
                           const float* __restrict__ xz,
                           const float* __restrict__ A_log,
                           const float* __restrict__ Dvec,
                           const float* __restrict__ Hinit,
                           float* __restrict__ Y)
{
    const int chunk = blockIdx.x;
    const int d = threadIdx.x;
    const int b = chunk / NCH;
    const int c = chunk % NCH;
    const int l0 = c * LC;

    float Av[DSTATE], h[DSTATE];
    const int base = chunk * (DINNER * DSTATE) + d * DSTATE;
#pragma unroll
    for (int n = 0; n < DSTATE; ++n) {
        Av[n] = -__expf(A_log[d * DSTATE + n]);
        h[n] = Hinit[base + n];
    }
    const float Dd = Dvec[d];

    for (int l = l0; l < l0 + LC; ++l) {
        const size_t m = (size_t)b * LSEQ + l;
        __builtin_prefetch(dbc + (m + 1) * DBC_PAD + DTRANK, 0, 0);
        const float dl = delta[m * DINNER + d];
        const float xv = xc[m * DINNER + d];
        const float* Bm = dbc + m * DBC_PAD + DTRANK;
        const float* Cm = Bm + DSTATE;
        const float dlx = dl * xv;
        float y = 0.0f;
#pragma unroll
        for (int n = 0; n < DSTATE; ++n) {
            const float a = __expf(dl * Av[n]);
            h[n] = fmaf(a, h[n], dlx * Bm[n]);
            y = fmaf(h[n], Cm[n], y);
        }
        y += xv * Dd;
        const float z = xz[m * DXZ + DINNER + d];
        const float sz = z / (1.0f + __expf(-z));
        Y[m * DINNER + d] = y * sz;
    }
}

// ---------------------------------------------------------------------------
extern "C" void kernel_launch(void* const* d_in, const int* in_sizes, int n_in,
                              void* d_out, int out_size, void* d_ws, size_t ws_size,
                              hipStream_t stream)
{
    const float* x      = (const float*)d_in[0];  // (B,H,W,96)
    const float* W_in   = (const float*)d_in[1];  // (384,96)
    const float* conv_w = (const float*)d_in[2];  // (192,3)
    const float* conv_b = (const float*)d_in[3];  // (192,)
    const float* W_x    = (const float*)d_in[4];  // (38,192)
    const float* W_dt   = (const float*)d_in[5];  // (192,6)
    const float* b_dt   = (const float*)d_in[6];  // (192,)
    const float* A_log  = (const float*)d_in[7];  // (192,16)
    const float* Dv     = (const float*)d_in[8];  // (192,)
    const float* W_out  = (const float*)d_in[9];  // (96,192)
    float* out = (float*)d_out;

    // workspace carve-up (floats)
    float* ws    = (float*)d_ws;
    float* xz    = ws;                                             // 18432*384
    float* xc    = xz    + (size_t)MTOT * DXZ;                     // 18432*192
    float* dbc   = xc    + (size_t)MTOT * DINNER;                  // 18432*48
    float* delta = dbc   + (size_t)MTOT * DBC_PAD;                 // 18432*192
    float* Pws   = delta + (size_t)MTOT * DINNER;                  // 192*3072
    float* Hend  = Pws   + (size_t)BQ * NCH * DINNER * DSTATE;
    float* Hinit = Hend  + (size_t)BQ * NCH * DINNER * DSTATE;
    float* Y     = Hinit + (size_t)BQ * NCH * DINNER * DSTATE;     // 18432*192

    // 1) xz = x @ W_in^T   (M=18432, K=96, N=384)
    {
        dim3 grid(MTOT / 64, DXZ / 64);
        gemm_wmma_f32<<<grid, 256, 0, stream>>>(x, DMODEL, W_in, DMODEL, DXZ,
                                                xz, DXZ, DMODEL);
    }
    // 2) depthwise causal conv + SiLU
    {
        const int tot = MTOT * DINNER;
        conv_silu_kernel<<<(tot + 255) / 256, 256, 0, stream>>>(xz, conv_w, conv_b, xc);
    }
    // 3) dbc = xc @ W_x^T  (M=18432, K=192, N=38 padded to 48)
    {
        dim3 grid(MTOT / 64, 1);
        gemm_wmma_f32<<<grid, 192, 0, stream>>>(xc, DINNER, W_x, DINNER, DBC_DIM,
                                                dbc, DBC_PAD, DINNER);
    }
    // 4) delta = softplus(dt @ W_dt^T + b_dt)
    {
        const int tot = MTOT * DINNER;
        delta_kernel<<<(tot + 255) / 256, 256, 0, stream>>>(dbc, W_dt, b_dt, delta);
    }
    // 5-7) chunked parallel scan
    scan_pass1<<<BQ * NCH, DINNER, 0, stream>>>(delta, dbc, xc, A_log, Pws, Hend);
    {
        const int tot = BQ * DINNER * DSTATE;
        scan_combine<<<(tot + 255) / 256, 256, 0, stream>>>(Pws, Hend, Hinit);
    }
    scan_pass2<<<BQ * NCH, DINNER, 0, stream>>>(delta, dbc, xc, xz, A_log, Dv, Hinit, Y);
    // 8) out = Y @ W_out^T (M=18432, K=192, N=96)
    {
        dim3 grid(MTOT / 64, 2);
        gemm_wmma_f32<<<grid, 192, 0, stream>>>(Y, DINNER, W_out, DINNER, DMODEL,
                                                out, DMODEL, DINNER);
    }
}